// Filter_89945205113415
// MI455X (gfx1250) — compile-verified
//
#include <hip/hip_runtime.h>
#include <hip/hip_bf16.h>
#include <math.h>

typedef __attribute__((ext_vector_type(2))) float v2f;
typedef __attribute__((ext_vector_type(8))) float v8f;

#define SIGLEN   262144
#define NBATCH   32
#define NFRAMES  1025
#define NBANDS   65
#define KPAD     68          // K padded to multiple of 4 for 16x16x4 WMMA
#define FILT     128
#define HOPSZ    256
#define NROWS    (NBATCH*NFRAMES)   // 32800 GEMM rows
#define MTILES   (NROWS/16)         // 2050 (exact)
#define GWAVES   2                  // waves per GEMM block (2050/2 = 1025 exact)
#define TWO_PI   6.28318530717958647692f

// ---- workspace layout (units: 4-byte words) ----
// basis stored K-pair-interleaved: basisP[k>>1][n][k&1], so a B-fragment
// (two consecutive K for one n) is a single aligned 8-byte load.
#define WS_BASIS 0                          // KPAD*FILT floats (pairs layout, zero-padded)
#define WS_H     (WS_BASIS + KPAD*FILT)     // NROWS*FILT impulse responses
#define WS_PEAK  (WS_H + NROWS*FILT)        // NBATCH floats: max|x| per batch
#define WS_MAXA  (WS_PEAK + NBATCH)         // NBATCH uints : max|y| per batch (bit-encoded)

// Basis: h[b,f,n] = sum_k mag[b,f,k] * basis[k][n]
// basis[k][n] = (1/128)*c_k*(-1)^k*cos(2*pi*k*n/128) * 0.5*(1-cos(2*pi*n/128))
// c_0 = c_64 = 1, else 2.  (-1)^k absorbs the roll-by-64 of the irfft.
__global__ void k_init(float* ws) {
    int i = blockIdx.x * blockDim.x + threadIdx.x;
    if (i < KPAD * FILT) {
        int k = i / FILT, n = i % FILT;
        float v = 0.0f;
        if (k < NBANDS) {
            float ck  = (k == 0 || k == 64) ? 1.0f : 2.0f;
            float sgn = (k & 1) ? -1.0f : 1.0f;
            int   kn  = (k * n) & 127;                 // arg reduction mod 128
            float c   = __cosf(TWO_PI * (float)kn * (1.0f / 128.0f));
            float w   = 0.5f * (1.0f - __cosf(TWO_PI * (float)n * (1.0f / 128.0f)));
            v = (1.0f / 128.0f) * ck * sgn * c * w;
        }
        // pair-interleaved layout: dst = (k/2)*(FILT*2) + n*2 + (k&1)
        ws[WS_BASIS + (k >> 1) * (FILT * 2) + n * 2 + (k & 1)] = v;
    }
    if (i < NBATCH) {
        ws[WS_PEAK + i] = 0.0f;
        ((unsigned*)ws)[WS_MAXA + i] = 0u;             // re-init every call (deterministic)
    }
}

// per-batch max|x| (one block per batch -> plain store, deterministic), b128 loads
__global__ void k_peak(const float* __restrict__ x, float* __restrict__ ws) {
    __shared__ float red[256];
    const int b = blockIdx.x;
    const float4* xb = (const float4*)(x + (size_t)b * SIGLEN);
    float m = 0.0f;
    for (int i = threadIdx.x; i < SIGLEN / 4; i += 256) {
        float4 v = xb[i];
        m = fmaxf(m, fmaxf(fmaxf(fabsf(v.x), fabsf(v.y)), fmaxf(fabsf(v.z), fabsf(v.w))));
    }
    red[threadIdx.x] = m;
    __syncthreads();
    for (int s = 128; s > 0; s >>= 1) {
        if (threadIdx.x < s) red[threadIdx.x] = fmaxf(red[threadIdx.x], red[threadIdx.x + s]);
        __syncthreads();
    }
    if (threadIdx.x == 0) ws[WS_PEAK + b] = red[0];
}

// (2*sigmoid(a)^ln10 + 1e-7)*10 = 20*exp(-ln10*log(1+exp(-a))) + 1e-6
// -> 3 transcendental VALU ops (v_exp_f32 x2, v_log_f32), no divide, no pow chain.
__device__ __forceinline__ float magxf(float a) {
    float t = __expf(-a);
    return 20.0f * __expf(-2.302585092994046f * __logf(1.0f + t)) + 1e-6f;
}

// h = mag x basis via V_WMMA_F32_16X16X4_F32.
// GWAVES waves per block; each wave owns one 16-row strip (16x128 outputs, 8 n-tiles).
// A tile (16x68, transformed, zero-padded) staged in LDS -> branch-free k-loop.
// launch_bounds(64,1): keep the full VGPR file, 64 f32 accumulators + fragments
// must stay in registers (no scratch spills in the WMMA loop).
// #pragma unroll 1: prevent the x2 unroll that doubled B-fragment liveness.
__global__ void __launch_bounds__(GWAVES * 32, 1)
k_impulse_gemm(const float* __restrict__ amps, float* __restrict__ ws) {
    __shared__ float sha[GWAVES][16 * KPAD];           // 2 * 4352 B
    const float* basisP = ws + WS_BASIS;               // pair-interleaved
    float*       h      = ws + WS_H;
    const int wave = threadIdx.x >> 5;
    const int lane = threadIdx.x & 31;
    const int m0   = (blockIdx.x * GWAVES + wave) * 16;
    const int half = lane >> 4;                        // 0: K=(k0,k0+1), 1: K=(k0+2,k0+3)
    const int l    = lane & 15;
    float* sa = sha[wave];

    // stage + transform the A tile once (each element transformed exactly once)
    for (int idx = lane; idx < 16 * KPAD; idx += 32) {
        const int r = idx / KPAD, k = idx - r * KPAD;
        sa[idx] = (k < NBANDS)
                ? magxf(amps[(size_t)(m0 + r) * NBANDS + k])
                : 0.0f;
    }
    __syncthreads();

    v8f acc[8] = {};
    #pragma unroll 1
    for (int k0 = 0; k0 < KPAD; k0 += 4) {
        const int ka = k0 + half * 2;                  // even, <= 66
        // A fragment: one ds_load_b64 (8-byte aligned: 68*4*l ≡ 0 mod 8, ka even)
        const v2f a = *(const v2f*)&sa[l * KPAD + ka];
        const size_t pbase = (size_t)(ka >> 1) * (FILT * 2);
        #pragma unroll
        for (int nt = 0; nt < 8; ++nt) {
            const int n = nt * 16 + l;
            // B fragment: one global_load_b64 from the pair-interleaved basis
            const v2f bb = *(const v2f*)&basisP[pbase + (size_t)n * 2];
            acc[nt] = __builtin_amdgcn_wmma_f32_16x16x4_f32(
                false, a, false, bb, (short)0, acc[nt], false, false);
        }
    }
    // D layout: VGPR v, lanes 0-15 -> M=v, lanes 16-31 -> M=v+8; N=lane&15
    #pragma unroll
    for (int v = 0; v < 8; ++v) {
        const int row = m0 + v + half * 8;
        #pragma unroll
        for (int nt = 0; nt < 8; ++nt)
            h[(size_t)row * FILT + nt * 16 + l] = acc[nt][v];
    }
}

// Time-domain FIR + window + overlap-add collapsed into a gather:
// block = 256 outputs of one hop chunk c (frames f=c and f-1 both cover it).
// y[s] = sum_k x[s+64-k] * ( wA*h_f[k]*[k<=t+64] + wB*h_{f-1}[k]*[k>=t-191] )
__global__ void k_fir(const float* __restrict__ x, const float* __restrict__ ws,
                      float* __restrict__ out) {
    __shared__ float sx[384];
    __shared__ float shf[128];
    __shared__ float shp[128];
    const int b   = blockIdx.y;
    const int c   = blockIdx.x;          // chunk == frame index f
    const int tid = threadIdx.x;
    const float* xb = x + (size_t)b * SIGLEN;
    const float* h  = ws + WS_H;
    const int B0 = c * HOPSZ;

    // stage x[B0-63 .. B0+320] (zero outside the signal)
    for (int j = tid; j < 384; j += 256) {
        int i = B0 - 63 + j;
        sx[j] = (i >= 0 && i < SIGLEN) ? xb[i] : 0.0f;
    }
    // stage the two filters
    if (tid < 128) {
        shf[tid] = h[((size_t)(b * NFRAMES + c)) * FILT + tid];
    } else {
        int j = tid - 128;
        shp[j] = (c >= 1) ? h[((size_t)(b * NFRAMES + c - 1)) * FILT + j] : 0.0f;
    }
    if (tid == 0 && c + 1 < SIGLEN / HOPSZ)
        __builtin_prefetch(&h[((size_t)(b * NFRAMES + c + 1)) * FILT], 0, 0);
    __syncthreads();

    // np.hanning(512): 0.5 - 0.5*cos(2*pi*n/511)
    const float wA = 0.5f - 0.5f * __cosf(TWO_PI * (float)tid * (1.0f / 511.0f));
    const float wB = 0.5f - 0.5f * __cosf(TWO_PI * (float)(tid + 256) * (1.0f / 511.0f));
    const bool hasPrev = (c >= 1);

    float acc = 0.0f;
    #pragma unroll 4
    for (int k = 0; k < 128; ++k) {
        float g = 0.0f;
        if (k <= tid + 64)              g += wA * shf[k];   // frame f clip (m>=0)
        if (hasPrev && k >= tid - 191)  g += wB * shp[k];   // frame f-1 clip (m<512)
        acc = fmaf(sx[tid + 127 - k], g, acc);              // x index s+64-k
    }
    out[(size_t)b * SIGLEN + B0 + tid] = acc;
}

// per-batch max|y| via order-independent atomic umax (monotone for x>=0), b128 loads
__global__ void k_maxabs(const float* __restrict__ out, unsigned* __restrict__ maxa) {
    __shared__ float red[256];
    const int b = blockIdx.y;
    const float4* ob = (const float4*)(out + (size_t)b * SIGLEN);
    float m = 0.0f;
    for (int i = blockIdx.x * 256 + threadIdx.x; i < SIGLEN / 4; i += gridDim.x * 256) {
        float4 v = ob[i];
        m = fmaxf(m, fmaxf(fmaxf(fabsf(v.x), fabsf(v.y)), fmaxf(fabsf(v.z), fabsf(v.w))));
    }
    red[threadIdx.x] = m;
    __syncthreads();
    for (int s = 128; s > 0; s >>= 1) {
        if (threadIdx.x < s) red[threadIdx.x] = fmaxf(red[threadIdx.x], red[threadIdx.x + s]);
        __syncthreads();
    }
    if (threadIdx.x == 0) atomicMax(&maxa[b], __float_as_uint(red[0]));
}

__global__ void k_scale(float* __restrict__ out, const float* __restrict__ ws) {
    const int b = blockIdx.y;
    const float peak = ws[WS_PEAK + b];
    const float mx   = __uint_as_float(((const unsigned*)ws)[WS_MAXA + b]);
    const float f    = peak / mx;
    float4* ob = (float4*)(out + (size_t)b * SIGLEN);
    for (int i = blockIdx.x * 256 + threadIdx.x; i < SIGLEN / 4; i += gridDim.x * 256) {
        float4 v = ob[i];
        v.x *= f; v.y *= f; v.z *= f; v.w *= f;
        ob[i] = v;
    }
}

extern "C" void kernel_launch(void* const* d_in, const int* in_sizes, int n_in,
                              void* d_out, int out_size, void* d_ws, size_t ws_size,
                              hipStream_t stream) {
    const float* x    = (const float*)d_in[0];   // (32,1,262144)
    const float* amps = (const float*)d_in[1];   // (32,1025,65)
    float* out = (float*)d_out;                  // (32,1,262144)
    float* ws  = (float*)d_ws;

    k_init<<<(KPAD * FILT + 255) / 256, 256, 0, stream>>>(ws);
    k_peak<<<NBATCH, 256, 0, stream>>>(x, ws);
    k_impulse_gemm<<<MTILES / GWAVES, 32 * GWAVES, 0, stream>>>(amps, ws);
    k_fir<<<dim3(SIGLEN / HOPSZ, NBATCH), 256, 0, stream>>>(x, ws, out);
    k_maxabs<<<dim3(64, NBATCH), 256, 0, stream>>>(out, (unsigned*)ws + WS_MAXA);
    k_scale<<<dim3(64, NBATCH), 256, 0, stream>>>(out, ws);
}